// GCN_91302414779135
// MI455X (gfx1250) — compile-verified
//
#include <hip/hip_runtime.h>

// ---------------- problem constants (match reference) ----------------
#define N_NODES   65536
#define NPG       128
#define NGRAPH    512
#define E_EDGES   1048576
#define HID       128
#define OUT_DIM   10
#define N_LAYERS  4
#define LN_EPS    1e-5f

typedef float v2f __attribute__((ext_vector_type(2)));
typedef float v4f __attribute__((ext_vector_type(4)));
typedef float v8f __attribute__((ext_vector_type(8)));

// explicit hardware f32 atomic add (no-return form -> STOREcnt only)
__device__ __forceinline__ void atomic_add_f32_hw(float* p, float v) {
    asm volatile("global_atomic_add_f32 %0, %1, off" : : "v"(p), "v"(v) : "memory");
}

// ---------------- degree build ----------------
__global__ void k_deg_init(int* __restrict__ degcnt) {
    int i = blockIdx.x * 256 + threadIdx.x;
    if (i < N_NODES) degcnt[i] = 0;
}

__global__ void k_deg_count(const int* __restrict__ edge_index, int* __restrict__ degcnt) {
    int e = blockIdx.x * 256 + threadIdx.x;
    if (e < E_EDGES) {
        int d = edge_index[E_EDGES + e];   // dst row of COO
        atomicAdd(&degcnt[d], 1);
    }
}

__global__ void k_deg_finish(const int* __restrict__ degcnt,
                             float* __restrict__ dinv, float* __restrict__ invdeg) {
    int i = blockIdx.x * 256 + threadIdx.x;
    if (i < N_NODES) {
        float d = 1.0f + (float)degcnt[i];   // self loop + in-degree
        dinv[i]   = rsqrtf(d);
        invdeg[i] = 1.0f / d;
    }
}

// ---------------- embedding: h = cls_emb[x0] + nbr_emb[x1] ----------------
__global__ void k_embed(const int* __restrict__ x,
                        const float* __restrict__ cls_emb,
                        const float* __restrict__ nbr_emb,
                        float* __restrict__ h) {
    int t = blockIdx.x * 256 + threadIdx.x;      // N*32 threads, 4 feats each
    int n = t >> 5;
    int f = (t & 31) << 2;
    int c  = x[2 * n + 0];
    int nb = x[2 * n + 1];
    v4f a = *(const v4f*)(cls_emb + (size_t)c  * HID + f);
    v4f b = *(const v4f*)(nbr_emb + (size_t)nb * HID + f);
    *(v4f*)(h + (size_t)n * HID + f) = a + b;
}

// ---------------- WMMA f32 GEMM: hw = h @ W ; agg = hw*invdeg + bias ----------------
// grid = N/128 blocks, 256 threads (8 waves). Each wave owns 16 rows x 128 cols
// (8 accumulator tiles of 16x16), K=128 in 32 steps of V_WMMA_F32_16X16X4_F32.
// W is staged in LDS *transposed* (col-major) with an XOR swizzle on k that
// keeps (k,k+1) pairs contiguous, so every B fragment is one ds_load_b64
// landing directly in an even-aligned VGPR pair (no repack movs).
__launch_bounds__(256)
__global__ void k_gemm_fused(const float* __restrict__ h,
                             const float* __restrict__ W,      // [128 x 128] row-major (K x N)
                             const float* __restrict__ bias,   // [128]
                             const float* __restrict__ invdeg, // [N]
                             float* __restrict__ hw,
                             float* __restrict__ agg) {
    __shared__ float lWt[HID * HID];   // 64 KB, layout: lWt[n*128 + (k ^ ((n&31)<<1))]
    const int tid = threadIdx.x;

    // stage W transposed+swizzled into LDS (global reads fully coalesced)
    for (int i = tid * 4; i < HID * HID; i += 256 * 4) {
        v4f w = *(const v4f*)(W + i);          // elements (k, n..n+3)
        const int k = i >> 7;
        const int n = i & (HID - 1);
#pragma unroll
        for (int j = 0; j < 4; ++j)
            lWt[(n + j) * HID + (k ^ (((n + j) & 31) << 1))] = w[j];
    }
    __syncthreads();

    const int wave = tid >> 5;
    const int lane = tid & 31;
    const int half = lane >> 4;     // 0: K pair {0,1}, 1: K pair {2,3} within a K4 step
    const int l16  = lane & 15;
    const int m0   = blockIdx.x * 128 + wave * 16;

    v8f zero = {};
    v8f acc[8];
#pragma unroll
    for (int t = 0; t < 8; ++t) acc[t] = zero;

    // A fragment rows: M = lane%16 (same for both lane halves)
    const float* arow = h + (size_t)(m0 + l16) * HID;

    for (int k = 0; k < HID; k += 4) {
        const int kb = k + half * 2;
        // A 16x4 f32 fragment: VGPR j holds K = kb + j (contiguous -> 8B load)
        v2f a = *(const v2f*)(arow + kb);
#pragma unroll
        for (int t = 0; t < 8; ++t) {
            const int col = t * 16 + l16;
            // B 4x16 f32 fragment: VGPR j holds K = kb + j for column N = lane%16
            v2f b = *(const v2f*)(lWt + col * HID + (kb ^ ((col & 31) << 1)));
            acc[t] = __builtin_amdgcn_wmma_f32_16x16x4_f32(
                false, a, false, b, (short)0, acc[t], false, false);
        }
    }

    // epilogue: C layout -> VGPR r holds M = r + 8*half, N = lane%16
    const int rbase = half * 8;
#pragma unroll
    for (int r = 0; r < 8; ++r) {
        const int row = m0 + rbase + r;
        const float inv = invdeg[row];
        float* hwrow  = hw  + (size_t)row * HID;
        float* aggrow = agg + (size_t)row * HID;
#pragma unroll
        for (int t = 0; t < 8; ++t) {
            const int col = t * 16 + l16;
            const float v = acc[t][r];
            hwrow[col]  = v;
            aggrow[col] = v * inv + bias[col];   // self-loop term + bias
        }
    }
}

// ---------------- edge scatter: agg[dst] += hw[src] * dinv[src]*dinv[dst] ----------------
// one wave per edge, 4 features per lane, hardware f32 atomics into L2
__global__ void k_scatter(const int* __restrict__ edge_index,
                          const float* __restrict__ dinv,
                          const float* __restrict__ hw,
                          float* __restrict__ agg) {
    int t = blockIdx.x * 256 + threadIdx.x;   // E*32 threads
    int e = t >> 5;
    if (e >= E_EDGES) return;
    int f = (t & 31) << 2;
    int s = edge_index[e];
    int d = edge_index[E_EDGES + e];
    float en = dinv[s] * dinv[d];
    v4f v = *(const v4f*)(hw + (size_t)s * HID + f);
    float* out = agg + (size_t)d * HID + f;
    atomic_add_f32_hw(out + 0, v.x * en);
    atomic_add_f32_hw(out + 1, v.y * en);
    atomic_add_f32_hw(out + 2, v.z * en);
    atomic_add_f32_hw(out + 3, v.w * en);
}

// ---------------- h = LayerNorm(h + relu(agg)) * gamma + beta, in place ----------------
// one wave32 per node: 4 features per lane, shfl_xor reductions
__launch_bounds__(256)
__global__ void k_relu_res_ln(float* __restrict__ h,
                              const float* __restrict__ agg,
                              const float* __restrict__ gamma,
                              const float* __restrict__ beta) {
    int t = blockIdx.x * 256 + threadIdx.x;
    int n = t >> 5;
    int lane = t & 31;
    int f = lane << 2;

    const float* ar = agg + (size_t)n * HID;
    float* hr = h + (size_t)n * HID;

    v4f hv = *(const v4f*)(hr + f);
    v4f av = *(const v4f*)(ar + f);
    v4f val;
    val.x = hv.x + fmaxf(av.x, 0.0f);
    val.y = hv.y + fmaxf(av.y, 0.0f);
    val.z = hv.z + fmaxf(av.z, 0.0f);
    val.w = hv.w + fmaxf(av.w, 0.0f);

    float s = val.x + val.y + val.z + val.w;
#pragma unroll
    for (int off = 16; off >= 1; off >>= 1) s += __shfl_xor(s, off, 32);
    const float mu = s * (1.0f / (float)HID);

    v4f dv = val - mu;
    float q = dv.x * dv.x + dv.y * dv.y + dv.z * dv.z + dv.w * dv.w;
#pragma unroll
    for (int off = 16; off >= 1; off >>= 1) q += __shfl_xor(q, off, 32);
    const float rstd = rsqrtf(q * (1.0f / (float)HID) + LN_EPS);

    v4f g = *(const v4f*)(gamma + f);
    v4f b = *(const v4f*)(beta + f);
    v4f outv = dv * rstd * g + b;
    *(v4f*)(hr + f) = outv;   // wave owns the whole row: in-place safe
}

// ---------------- logits: out[g] = h[root(g)] @ W_out ----------------
__global__ void k_out(const float* __restrict__ h,
                      const float* __restrict__ W_out,
                      float* __restrict__ out) {
    int t = blockIdx.x * 256 + threadIdx.x;
    if (t >= NGRAPH * OUT_DIM) return;
    int g = t / OUT_DIM;
    int o = t % OUT_DIM;
    const float* hr = h + (size_t)(g * NPG) * HID;   // root = first node of graph
    float s = 0.0f;
#pragma unroll 4
    for (int k = 0; k < HID; ++k) s = fmaf(hr[k], W_out[k * OUT_DIM + o], s);
    out[t] = s;
}

// ---------------- launch ----------------
extern "C" void kernel_launch(void* const* d_in, const int* in_sizes, int n_in,
                              void* d_out, int out_size, void* d_ws, size_t ws_size,
                              hipStream_t stream) {
    const int*   x        = (const int*)d_in[0];
    const int*   ei       = (const int*)d_in[1];
    // d_in[2] = batch (unused: graphs are uniform, root = g*NPG)
    const float* cls_emb  = (const float*)d_in[3];
    const float* nbr_emb  = (const float*)d_in[4];
    const float* Ws       = (const float*)d_in[5];
    const float* bs       = (const float*)d_in[6];
    const float* gammas   = (const float*)d_in[7];
    const float* betas    = (const float*)d_in[8];
    const float* W_out    = (const float*)d_in[9];
    float*       out      = (float*)d_out;

    // workspace carve-up (~96.8 MB needed)
    char* ws = (char*)d_ws;
    size_t off = 0;
    int*   degcnt = (int*)(ws + off);   off += (size_t)N_NODES * 4;
    float* dinv   = (float*)(ws + off); off += (size_t)N_NODES * 4;
    float* invdeg = (float*)(ws + off); off += (size_t)N_NODES * 4;
    float* h      = (float*)(ws + off); off += (size_t)N_NODES * HID * 4;
    float* hw     = (float*)(ws + off); off += (size_t)N_NODES * HID * 4;
    float* agg    = (float*)(ws + off); off += (size_t)N_NODES * HID * 4;
    (void)ws_size; (void)in_sizes; (void)n_in; (void)out_size;

    k_deg_init  <<<N_NODES / 256, 256, 0, stream>>>(degcnt);
    k_deg_count <<<E_EDGES / 256, 256, 0, stream>>>(ei, degcnt);
    k_deg_finish<<<N_NODES / 256, 256, 0, stream>>>(degcnt, dinv, invdeg);

    k_embed<<<(N_NODES * 32) / 256, 256, 0, stream>>>(x, cls_emb, nbr_emb, h);

    for (int i = 0; i < N_LAYERS; ++i) {
        k_gemm_fused<<<N_NODES / 128, 256, 0, stream>>>(
            h, Ws + (size_t)i * HID * HID, bs + (size_t)i * HID, invdeg, hw, agg);
        k_scatter<<<(E_EDGES * 32) / 256, 256, 0, stream>>>(ei, dinv, hw, agg);
        k_relu_res_ln<<<(N_NODES * 32) / 256, 256, 0, stream>>>(
            h, agg, gammas + (size_t)i * HID, betas + (size_t)i * HID);
    }

    k_out<<<(NGRAPH * OUT_DIM + 255) / 256, 256, 0, stream>>>(h, W_out, out);
}